// CNN_1D_quantizer_QE_12128987644311
// MI455X (gfx1250) — compile-verified
//
#include <hip/hip_runtime.h>
#include <hip/hip_bf16.h>
#include <stdint.h>

// ---------------------------------------------------------------------------
// Euclidean VQ for MI455X (gfx1250):
//   dist = 2*z@C^T - |z|^2 - |c|^2   via V_WMMA_F32_16X16X4_F32
//   global->LDS staging via GLOBAL_LOAD_ASYNC_TO_LDS_B128 (ASYNCcnt)
//   argmax per row via packed-u64 global atomic max
//   zq gather + straight-through + commitment loss
// ---------------------------------------------------------------------------

typedef __attribute__((ext_vector_type(2))) float v2f;
typedef __attribute__((ext_vector_type(8))) float v8f;

#define BN_ROWS 16384   // B*N = 8*2048
#define D_DIM   256
#define KCODES  8192

// d_out layout (floats), outputs concatenated in return order
#define ZQ_OFF   0
#define IDX_OFF  4194304     // 8*2048*256
#define LOSS_OFF 4210688     // + 16384
#define DIST_OFF 4210689     // + 1

#define LDST 68              // padded LDS row stride in floats (bank-conflict free)

// CDNA5 async global->LDS copy path (device pass only; host pass parses fallback)
#if defined(__gfx1250__) && __has_builtin(__builtin_amdgcn_global_load_async_to_lds_b128)
#define HAS_ASYNC_LDS 1
typedef __attribute__((ext_vector_type(4))) int v4i;
typedef __attribute__((address_space(1))) v4i g_v4i;   // global int4
typedef __attribute__((address_space(3))) v4i l_v4i;   // LDS int4

__device__ __forceinline__ void async_copy_b128(const float* gsrc, float* lds) {
  __builtin_amdgcn_global_load_async_to_lds_b128(
      (g_v4i*)(v4i*)const_cast<float*>(gsrc),
      (l_v4i*)(v4i*)lds, 0, 0);
}
#endif

__device__ __forceinline__ void wait_async_then_barrier() {
#if defined(HAS_ASYNC_LDS)
#if __has_builtin(__builtin_amdgcn_s_wait_asynccnt)
  __builtin_amdgcn_s_wait_asynccnt(0);
#else
  asm volatile("s_wait_asynccnt 0x0" ::: "memory");
#endif
#endif
  __syncthreads();
}

__device__ __forceinline__ unsigned long long shfl_xor_u64(unsigned long long v, int m) {
  int lo = __shfl_xor((int)(unsigned int)v, m, 32);
  int hi = __shfl_xor((int)(unsigned int)(v >> 32), m, 32);
  return ((unsigned long long)(unsigned int)hi << 32) | (unsigned int)lo;
}

// monotone float -> u32 key (total order, negatives below positives)
__device__ __forceinline__ unsigned int fkey(float f) {
  unsigned int b = __float_as_uint(f);
  return (b & 0x80000000u) ? ~b : (b | 0x80000000u);
}

// ---------------------------------------------------------------------------
__global__ void vq_init_kernel(unsigned long long* __restrict__ argp,
                               float* __restrict__ loss) {
  int i = blockIdx.x * blockDim.x + threadIdx.x;
  if (i < BN_ROWS) argp[i] = 0ull;
  if (i == 0) loss[0] = 0.0f;
}

// ---------------------------------------------------------------------------
// one wave32 per row: ||z_i||^2 for rows < BN_ROWS, else ||c_k||^2
__global__ __launch_bounds__(256) void vq_norms_kernel(const float* __restrict__ z,
                                                       const float* __restrict__ cb,
                                                       float* __restrict__ znorm,
                                                       float* __restrict__ cnorm) {
  int wid  = (blockIdx.x * blockDim.x + threadIdx.x) >> 5;
  int lane = threadIdx.x & 31;
  const float4* src;
  float* dst;
  if (wid < BN_ROWS) {
    src = (const float4*)(z + (size_t)wid * D_DIM);
    dst = znorm + wid;
  } else {
    int r = wid - BN_ROWS;
    src = (const float4*)(cb + (size_t)r * D_DIM);
    dst = cnorm + r;
  }
  float s = 0.0f;
  #pragma unroll
  for (int i = lane; i < D_DIM / 4; i += 32) {
    float4 v = src[i];
    s += v.x * v.x + v.y * v.y + v.z * v.z + v.w * v.w;
  }
  #pragma unroll
  for (int m = 16; m; m >>= 1) s += __shfl_xor(s, m, 32);
  if (lane == 0) *dst = s;
}

// ---------------------------------------------------------------------------
// GEMM: block tile 64(M) x 128(N), K chunked by 64 through LDS.
// 8 waves: 4 (M) x 2 (N); each wave: 16x64 tile = 4 v8f accumulators.
__global__ __launch_bounds__(256) void vq_gemm_kernel(const float* __restrict__ z,
                                                      const float* __restrict__ cb,
                                                      const float* __restrict__ znorm,
                                                      const float* __restrict__ cnorm,
                                                      unsigned long long* __restrict__ argp,
                                                      float* __restrict__ dist_out) {
  __shared__ float As[64 * LDST];    // 17408 B
  __shared__ float Bs[128 * LDST];   // 34816 B

  const int m_base = blockIdx.y * 64;
  const int n_base = blockIdx.x * 128;
  const int tid    = threadIdx.x;
  const int wave   = tid >> 5;
  const int lane   = tid & 31;
  const int lane16 = lane & 15;
  const int hi     = lane >> 4;            // which half of the wave
  const int m_off  = (wave & 3) * 16;      // wave's M offset in block tile
  const int n_off  = (wave >> 2) * 64;     // wave's N offset in block tile

  v8f zero = {0.f, 0.f, 0.f, 0.f, 0.f, 0.f, 0.f, 0.f};
  v8f acc[4];
  acc[0] = zero; acc[1] = zero; acc[2] = zero; acc[3] = zero;

  const float* Abase = As + (m_off + lane16) * LDST + 2 * hi;
  const float* Bbase = Bs + (n_off + lane16) * LDST + 2 * hi;

  for (int kc = 0; kc < D_DIM / 64; ++kc) {
    const int kbase = kc * 64;

#if defined(HAS_ASYNC_LDS)
    // ---- async global->LDS staging: no VGPR round trip, ASYNCcnt tracked ----
    #pragma unroll
    for (int i = 0; i < 4; ++i) {
      int id  = tid + 256 * i;
      int row = id >> 4;
      int q   = id & 15;
      async_copy_b128(z + (size_t)(m_base + row) * D_DIM + kbase + q * 4,
                      As + row * LDST + q * 4);
    }
    #pragma unroll
    for (int i = 0; i < 8; ++i) {
      int id  = tid + 256 * i;
      int row = id >> 4;
      int q   = id & 15;
      async_copy_b128(cb + (size_t)(n_base + row) * D_DIM + kbase + q * 4,
                      Bs + row * LDST + q * 4);
    }
#else
    // ---- fallback: distinct temporaries so loads can overlap ----
    float4 ta[4];
    float4 tb[8];
    #pragma unroll
    for (int i = 0; i < 4; ++i) {
      int id = tid + 256 * i;
      ta[i] = *(const float4*)(z + (size_t)(m_base + (id >> 4)) * D_DIM + kbase + (id & 15) * 4);
    }
    #pragma unroll
    for (int i = 0; i < 8; ++i) {
      int id = tid + 256 * i;
      tb[i] = *(const float4*)(cb + (size_t)(n_base + (id >> 4)) * D_DIM + kbase + (id & 15) * 4);
    }
    #pragma unroll
    for (int i = 0; i < 4; ++i) {
      int id = tid + 256 * i;
      *(float4*)(As + (id >> 4) * LDST + (id & 15) * 4) = ta[i];
    }
    #pragma unroll
    for (int i = 0; i < 8; ++i) {
      int id = tid + 256 * i;
      *(float4*)(Bs + (id >> 4) * LDST + (id & 15) * 4) = tb[i];
    }
#endif
    wait_async_then_barrier();

    // ---- 16 WMMA K-steps of 4, register double-buffered fragments ----
    v2f a[2];
    v2f b[2][4];
    a[0] = *(const v2f*)(Abase);
    #pragma unroll
    for (int t = 0; t < 4; ++t) b[0][t] = *(const v2f*)(Bbase + t * 16 * LDST);

    #pragma unroll
    for (int ks = 0; ks < 16; ++ks) {
      const int cur = ks & 1;
      const int nxt = cur ^ 1;
      if (ks < 15) {
        const int k0 = (ks + 1) * 4;
        a[nxt] = *(const v2f*)(Abase + k0);
        #pragma unroll
        for (int t = 0; t < 4; ++t)
          b[nxt][t] = *(const v2f*)(Bbase + t * 16 * LDST + k0);
      }
      #pragma unroll
      for (int t = 0; t < 4; ++t) {
        acc[t] = __builtin_amdgcn_wmma_f32_16x16x4_f32(
            /*neg_a=*/false, a[cur], /*neg_b=*/false, b[cur][t],
            /*c_mod=*/(short)0, acc[t], /*reuse_a=*/false, /*reuse_b=*/false);
      }
    }
    __syncthreads();
  }

  // ---- epilogue: dist = 2*acc - |z|^2 - |c|^2, streamed NT; per-row argmax ----
  float cn[4];
  #pragma unroll
  for (int t = 0; t < 4; ++t) cn[t] = cnorm[n_base + n_off + t * 16 + lane16];
  float zn[8];
  #pragma unroll
  for (int j = 0; j < 8; ++j) zn[j] = znorm[m_base + m_off + hi * 8 + j];

  #pragma unroll
  for (int j = 0; j < 8; ++j) {
    const int grow = m_base + m_off + hi * 8 + j;   // global row for this VGPR/half
    unsigned long long best = 0ull;
    #pragma unroll
    for (int t = 0; t < 4; ++t) {
      const int gcol = n_base + n_off + t * 16 + lane16;
      float dv = 2.0f * acc[t][j] - zn[j] - cn[t];
      // dist section of d_out is only 4-byte aligned -> b32 NT stores
      __builtin_nontemporal_store(dv, dist_out + (size_t)grow * KCODES + gcol);
      unsigned long long pk =
          ((unsigned long long)fkey(dv) << 32) | (unsigned int)(~(unsigned int)gcol);
      best = (pk > best) ? pk : best;
    }
    // reduce across the 16 lanes sharing this row (stay within half-wave)
    #pragma unroll
    for (int m = 1; m <= 8; m <<= 1) {
      unsigned long long o = shfl_xor_u64(best, m);
      if (o > best) best = o;
    }
    if (lane16 == 0) atomicMax(argp + grow, best);
  }
}

// ---------------------------------------------------------------------------
// one block per row: gather zq, straight-through value, index, loss partial
__global__ __launch_bounds__(256) void vq_gather_kernel(const float* __restrict__ z,
                                                        const float* __restrict__ cb,
                                                        const unsigned long long* __restrict__ argp,
                                                        float* __restrict__ out,
                                                        float* __restrict__ loss_sum) {
  const int row = blockIdx.x;
  const int t   = threadIdx.x;
  unsigned long long p = argp[row];
  unsigned int idx = ~(unsigned int)(p & 0xFFFFFFFFull);

  float zv = z[(size_t)row * D_DIM + t];
  float cv = cb[(size_t)idx * D_DIM + t];
  float q  = zv + (cv - zv);                 // straight-through forward value
  out[ZQ_OFF + (size_t)row * D_DIM + t] = q;
  if (t == 0) out[IDX_OFF + row] = (float)idx;

  float d  = cv - zv;
  float sq = d * d;
  __shared__ float red[8];
  #pragma unroll
  for (int m = 16; m; m >>= 1) sq += __shfl_xor(sq, m, 32);
  int w = t >> 5, lane = t & 31;
  if (lane == 0) red[w] = sq;
  __syncthreads();
  if (w == 0) {
    float v = (lane < 8) ? red[lane] : 0.0f;
    #pragma unroll
    for (int m = 4; m; m >>= 1) v += __shfl_xor(v, m, 32);
    if (lane == 0) atomicAdd(loss_sum, v);
  }
}

// ---------------------------------------------------------------------------
__global__ void vq_finalize_kernel(const float* __restrict__ loss_sum,
                                   float* __restrict__ out) {
  out[LOSS_OFF] = loss_sum[0] * (1.0f / ((float)BN_ROWS * (float)D_DIM));
}

// ---------------------------------------------------------------------------
extern "C" void kernel_launch(void* const* d_in, const int* in_sizes, int n_in,
                              void* d_out, int out_size, void* d_ws, size_t ws_size,
                              hipStream_t stream) {
  const float* z  = (const float*)d_in[0];   // [8,2048,256] f32
  const float* cb = (const float*)d_in[1];   // [8192,256]   f32
  float* out = (float*)d_out;

  char* ws = (char*)d_ws;
  unsigned long long* argp = (unsigned long long*)ws;                     // 16384*8 B
  float* znorm = (float*)(ws + (size_t)BN_ROWS * 8);                      // 16384*4 B
  float* cnorm = (float*)(ws + (size_t)BN_ROWS * 8 + (size_t)BN_ROWS * 4);// 8192*4 B
  float* loss  = (float*)(ws + (size_t)BN_ROWS * 8 + (size_t)BN_ROWS * 4
                             + (size_t)KCODES * 4);                       // 4 B

  vq_init_kernel<<<BN_ROWS / 256, 256, 0, stream>>>(argp, loss);
  vq_norms_kernel<<<(BN_ROWS + KCODES) / 8, 256, 0, stream>>>(z, cb, znorm, cnorm);
  vq_gemm_kernel<<<dim3(KCODES / 128, BN_ROWS / 64), 256, 0, stream>>>(
      z, cb, znorm, cnorm, argp, out + DIST_OFF);
  vq_gather_kernel<<<BN_ROWS, 256, 0, stream>>>(z, cb, argp, out, loss);
  vq_finalize_kernel<<<1, 1, 0, stream>>>(loss, out);
}